// MultiNormReconstructionLoss_58617713656349
// MI455X (gfx1250) — compile-verified
//
#include <hip/hip_runtime.h>

typedef float v2f __attribute__((ext_vector_type(2)));
typedef float v4f __attribute__((ext_vector_type(4)));
typedef float v8f __attribute__((ext_vector_type(8)));

#define ROWS     16
#define NCOL     1048576
#define BINS     1024
#define SEGS     16                    // blocks per row -> 256 blocks total
#define THREADS  256
#define ITERS    ((NCOL / SEGS) / (THREADS * 4))   // 64 float4 iterations/thread
#define NREP     8                     // one LDS histogram replica per wave
#define HSTRIDE  (BINS + 1)            // +1 so replicas hit different LDS banks
#define K_TOP    2048

// ---------------------------------------------------------------- zero ws
__global__ __launch_bounds__(THREADS)
void zero_ws_kernel(unsigned* __restrict__ p, int n) {
    for (int i = blockIdx.x * THREADS + threadIdx.x; i < n; i += gridDim.x * THREADS)
        p[i] = 0u;
}

// ------------------------------------------------- fused streaming pass
// One read of y/yh/mask. Produces:
//   gcnt/gsum : per-row 1024-bin histogram (count, sum) of d^2  (atomics)
//   blk_abs/blk_sq : per-block partial sums of |d| and d^2      (deterministic)
__global__ __launch_bounds__(THREADS)
void fused_pass(const v4f* __restrict__ y, const v4f* __restrict__ yh,
                const v4f* __restrict__ mk,
                unsigned* __restrict__ gcnt, float* __restrict__ gsum,
                float* __restrict__ blk_abs, float* __restrict__ blk_sq)
{
    __shared__ unsigned hc[NREP][HSTRIDE];
    __shared__ float    hs[NREP][HSTRIDE];
    __shared__ float    red[2][THREADS];

    const int tid  = threadIdx.x;
    const int wave = tid >> 5;

    for (int i = tid; i < NREP * HSTRIDE; i += THREADS) {
        (&hc[0][0])[i] = 0u;
        (&hs[0][0])[i] = 0.0f;
    }
    __syncthreads();

    const int row = blockIdx.y, seg = blockIdx.x;
    const size_t base4 = (size_t)row * (NCOL / 4) + (size_t)seg * (NCOL / (SEGS * 4));

    float s_abs = 0.0f, s_sq = 0.0f;
    for (int it = 0; it < ITERS; ++it) {
        const size_t idx = base4 + (size_t)it * THREADS + tid;
        v4f a = __builtin_nontemporal_load(&y[idx]);
        v4f b = __builtin_nontemporal_load(&yh[idx]);
        v4f w = __builtin_nontemporal_load(&mk[idx]);
        v4f d = (a - b) * w;
#pragma unroll
        for (int c = 0; c < 4; ++c) {
            float dv = d[c];
            float d2 = dv * dv;
            s_abs += __builtin_fabsf(dv);
            s_sq  += d2;
            unsigned bin = __float_as_uint(d2) >> 21;   // sign(0)+exp8+mant2 -> 0..1023
            atomicAdd(&hc[wave][bin], 1u);
            atomicAdd(&hs[wave][bin], d2);
        }
    }
    __syncthreads();

    // merge 8 replicas, push to global per-row histogram
    unsigned* gc = gcnt + (size_t)row * BINS;
    float*    gs = gsum + (size_t)row * BINS;
    for (int b = tid; b < BINS; b += THREADS) {
        unsigned c = 0; float s = 0.0f;
#pragma unroll
        for (int r = 0; r < NREP; ++r) { c += hc[r][b]; s += hs[r][b]; }
        if (c) { atomicAdd(&gc[b], c); atomicAdd(&gs[b], s); }
    }

    // ---- block reduction of (s_abs, s_sq) via V_WMMA_F32_16X16X4_F32 ----
    red[0][tid] = s_abs;
    red[1][tid] = s_sq;
    __syncthreads();
    if (tid < 32) {                      // wave 0, EXEC all ones
        v2f pa, pq, ones;
        pa.x = red[0][tid]      + red[0][tid + 64]  + red[0][tid + 128] + red[0][tid + 192];
        pa.y = red[0][tid + 32] + red[0][tid + 96]  + red[0][tid + 160] + red[0][tid + 224];
        pq.x = red[1][tid]      + red[1][tid + 64]  + red[1][tid + 128] + red[1][tid + 192];
        pq.y = red[1][tid + 32] + red[1][tid + 96]  + red[1][tid + 160] + red[1][tid + 224];
        ones.x = 1.0f; ones.y = 1.0f;
        v8f cz = {};
        // A = 16x4 partials, B = 4x16 ones => D[m][*] = rowsum(A,m)
        v8f da = __builtin_amdgcn_wmma_f32_16x16x4_f32(
            false, pa, false, ones, (short)0, cz, false, false);
        v8f dq = __builtin_amdgcn_wmma_f32_16x16x4_f32(
            false, pq, false, ones, (short)0, cz, false, false);
        // lanes 0-15 hold s0..s7 in c[0..7]; lanes 16-31 hold s8..s15
        float ta = da[0] + da[1] + da[2] + da[3] + da[4] + da[5] + da[6] + da[7];
        float tq = dq[0] + dq[1] + dq[2] + dq[3] + dq[4] + dq[5] + dq[6] + dq[7];
        ta += __shfl_xor(ta, 16, 32);
        tq += __shfl_xor(tq, 16, 32);
        if (tid == 0) {
            const int bid = row * SEGS + seg;
            blk_abs[bid] = ta;
            blk_sq[bid]  = tq;
        }
    }
}

// --------------------------------------- histogram selection + finalize
__global__ __launch_bounds__(THREADS)
void select_finalize(const unsigned* __restrict__ gcnt, const float* __restrict__ gsum,
                     const float* __restrict__ blk_abs, const float* __restrict__ blk_sq,
                     float* __restrict__ out)
{
    __shared__ float sc[BINS], ss[BINS], tc[BINS], ts[BINS];
    __shared__ int   bstar;
    __shared__ float acc;
    __shared__ float red[THREADS];

    const int tid = threadIdx.x;
    if (tid == 0) acc = 0.0f;

    for (int row = 0; row < ROWS; ++row) {
        for (int b = tid; b < BINS; b += THREADS) {
            sc[b] = (float)gcnt[row * BINS + b];    // counts <= 2^20, exact in f32
            ss[b] = gsum[row * BINS + b];
        }
        __syncthreads();
        // inclusive suffix scan (Hillis-Steele, double-buffered)
        for (int off = 1; off < BINS; off <<= 1) {
            for (int b = tid; b < BINS; b += THREADS) {
                float c = sc[b], s = ss[b];
                if (b + off < BINS) { c += sc[b + off]; s += ss[b + off]; }
                tc[b] = c; ts[b] = s;
            }
            __syncthreads();
            for (int b = tid; b < BINS; b += THREADS) { sc[b] = tc[b]; ss[b] = ts[b]; }
            __syncthreads();
        }
        const float K = (float)K_TOP;
        for (int b = tid; b < BINS; b += THREADS) {
            float here = sc[b];
            float nxt  = (b + 1 < BINS) ? sc[b + 1] : 0.0f;
            if (here >= K && nxt < K) bstar = b;     // unique crossing bin
        }
        __syncthreads();
        if (tid == 0) {
            int   b       = bstar;
            float above_c = (b + 1 < BINS) ? sc[b + 1] : 0.0f;
            float above_s = (b + 1 < BINS) ? ss[b + 1] : 0.0f;
            float bin_c   = sc[b] - above_c;
            float bin_s   = ss[b] - above_s;
            float need    = K - above_c;
            float row_top = above_s + ((bin_c > 0.0f) ? need * (bin_s / bin_c) : 0.0f);
            acc += row_top;
        }
        __syncthreads();
    }

    // deterministic reduction of the 256 per-block partials
    float va = blk_abs[tid];
    float vq = blk_sq[tid];
    red[tid] = vq; __syncthreads();
    for (int off = THREADS / 2; off > 0; off >>= 1) {
        if (tid < off) red[tid] += red[tid + off];
        __syncthreads();
    }
    float total_sq = red[0];
    __syncthreads();
    red[tid] = va; __syncthreads();
    for (int off = THREADS / 2; off > 0; off >>= 1) {
        if (tid < off) red[tid] += red[tid + off];
        __syncthreads();
    }
    float total_abs = red[0];

    if (tid == 0) {
        const float invN = 1.0f / ((float)ROWS * (float)NCOL);
        float l1   = total_abs * invN;
        float l2   = total_sq  * invN;
        float linf = acc / (float)ROWS;
        out[0] = 0.0f * l1 + 1.0f * l2 + 0.02f * linf;
    }
}

// ---------------------------------------------------------------- launch
extern "C" void kernel_launch(void* const* d_in, const int* in_sizes, int n_in,
                              void* d_out, int out_size, void* d_ws, size_t ws_size,
                              hipStream_t stream)
{
    const v4f* y  = (const v4f*)d_in[0];
    const v4f* yh = (const v4f*)d_in[1];
    const v4f* mk = (const v4f*)d_in[2];

    unsigned char* ws = (unsigned char*)d_ws;
    unsigned* gcnt    = (unsigned*)ws;                  // 16*1024*4  = 65536 B
    float*    gsum    = (float*)(ws + 65536);           // 16*1024*4  = 65536 B
    float*    blk_abs = (float*)(ws + 131072);          // 256*4
    float*    blk_sq  = (float*)(ws + 132096);          // 256*4

    // zero the atomically-accumulated histogram region every call
    zero_ws_kernel<<<64, THREADS, 0, stream>>>((unsigned*)ws, (BINS * ROWS * 2));

    dim3 grid(SEGS, ROWS);
    fused_pass<<<grid, THREADS, 0, stream>>>(y, yh, mk, gcnt, gsum, blk_abs, blk_sq);

    select_finalize<<<1, THREADS, 0, stream>>>(gcnt, gsum, blk_abs, blk_sq, (float*)d_out);
}